// QAttentionLayer_65481071399469
// MI455X (gfx1250) — compile-verified
//
#include <hip/hip_runtime.h>

// out[row, j] = prod_{k<=j} cos(x[row,k]) * cos(w[k])
//  = (cumprod_j cos(x[row,*])) * (cumprod_j cos(w[*]))   (products commute)
//
// Streaming, HBM-bound kernel (~336 MB traffic -> ~14.4 us floor @ 23.3 TB/s).
// Data path: async global->LDS B128 (coalesced, ASYNCcnt), ds_load_b64 row
// reads, v_cos_f32 + v_mul_f32, global_store_b64.

#define NQ 10
#define BLOCK 256
#define ROWS_PER_BLOCK (2 * BLOCK)                 // 512 rows / block
#define FLOATS_PER_BLOCK (ROWS_PER_BLOCK * NQ)     // 5120 floats = 20 KiB
#define VEC4_PER_BLOCK (FLOATS_PER_BLOCK / 4)      // 1280 float4
#define VEC4_PER_THREAD (VEC4_PER_BLOCK / BLOCK)   // 5 per thread

typedef int v4i __attribute__((ext_vector_type(4)));
typedef __attribute__((address_space(1))) v4i* global_v4i_p;
typedef __attribute__((address_space(3))) v4i* lds_v4i_p;

#if defined(__has_builtin)
#if __has_builtin(__builtin_amdgcn_global_load_async_to_lds_b128)
#define HAVE_ASYNC_BUILTIN 1
#endif
#if __has_builtin(__builtin_amdgcn_s_wait_asynccnt)
#define HAVE_WAIT_BUILTIN 1
#endif
#endif

__device__ __forceinline__ void async_g2l_b128(const float* g, float* l) {
#ifdef HAVE_ASYNC_BUILTIN
  __builtin_amdgcn_global_load_async_to_lds_b128(
      (global_v4i_p)g, (lds_v4i_p)l, /*imm offset*/ 0, /*cpol*/ 0);
#else
  // Generic LDS pointer: low 32 bits are the LDS byte offset.
  unsigned lds_off = (unsigned)(size_t)l;
  asm volatile("global_load_async_to_lds_b128 %0, %1, off"
               :
               : "v"(lds_off), "v"(g)
               : "memory");
#endif
}

__device__ __forceinline__ void wait_async0() {
#ifdef HAVE_WAIT_BUILTIN
  __builtin_amdgcn_s_wait_asynccnt(0);
#else
  asm volatile("s_wait_asynccnt 0" ::: "memory");
#endif
}

__global__ __launch_bounds__(BLOCK) void qcircuit_cumprod_kernel(
    const float* __restrict__ x, const float* __restrict__ w,
    float* __restrict__ out, unsigned total_rows) {
  __shared__ float lds[FLOATS_PER_BLOCK];
  const unsigned tid = threadIdx.x;
  const unsigned block_row0 = blockIdx.x * ROWS_PER_BLOCK;

  // Cumulative product of cos(weights): uniform scalar loads, 10 trans ops.
  float cwc[NQ];
  {
    float p = 1.0f;
#pragma unroll
    for (int j = 0; j < NQ; ++j) {
      p *= __cosf(w[j]);
      cwc[j] = p;
    }
  }

  if (block_row0 + ROWS_PER_BLOCK <= total_rows) {
    const unsigned base = block_row0 * NQ;  // flat float index of block start
    const float* gsrc = x + base;

    // Coalesced async copy of 20 KiB: 5 x b128 per lane, contiguous per wave.
#pragma unroll
    for (int k = 0; k < VEC4_PER_THREAD; ++k) {
      const unsigned fo = (tid + k * BLOCK) * 4;  // float offset, 16B aligned
      async_g2l_b128(gsrc + fo, lds + fo);
    }
    wait_async0();     // this wave's async loads landed in LDS
    __syncthreads();   // all waves' loads landed

    // Each thread owns rows (tid) and (tid + 256): lane stride of 10 dwords
    // in LDS -> 32 distinct banks across a wave32 (gcd(10,64)=2).
#pragma unroll
    for (int rr = 0; rr < 2; ++rr) {
      const unsigned row = tid + rr * BLOCK;
      const float2* lrow = (const float2*)(lds + row * NQ);      // 8B aligned
      float2* orow = (float2*)(out + base + row * NQ);           // 8B aligned
      float p = 1.0f;
#pragma unroll
      for (int k = 0; k < NQ / 2; ++k) {
        const float2 v = lrow[k];  // ds_load_b64
        float2 o;
        p *= __cosf(v.x);
        o.x = p * cwc[2 * k];
        p *= __cosf(v.y);
        o.y = p * cwc[2 * k + 1];
        orow[k] = o;               // global_store_b64
      }
    }
  } else {
    // Guarded scalar tail path (not taken for the reference shape).
    for (unsigned row = block_row0 + tid; row < total_rows; row += BLOCK) {
      const float* xr = x + (size_t)row * NQ;
      float* orow = out + (size_t)row * NQ;
      float p = 1.0f;
#pragma unroll
      for (int j = 0; j < NQ; ++j) {
        p *= __cosf(xr[j]);
        orow[j] = p * cwc[j];
      }
    }
  }
}

extern "C" void kernel_launch(void* const* d_in, const int* in_sizes, int n_in,
                              void* d_out, int out_size, void* d_ws, size_t ws_size,
                              hipStream_t stream) {
  const float* x = (const float*)d_in[0];   // (16384, 256, 10) fp32
  const float* w = (const float*)d_in[1];   // (10,) fp32
  float* out = (float*)d_out;               // (16384, 256, 10) fp32

  const unsigned total_rows = (unsigned)(in_sizes[0] / NQ);  // 4,194,304
  const unsigned blocks = (total_rows + ROWS_PER_BLOCK - 1) / ROWS_PER_BLOCK;

  qcircuit_cumprod_kernel<<<dim3(blocks), dim3(BLOCK), 0, stream>>>(
      x, w, out, total_rows);
}